// GTRTree_18245021073882
// MI455X (gfx1250) — compile-verified
//
#include <hip/hip_runtime.h>
#include <math.h>

// ---------------------------------------------------------------------------
// Phylogenetic likelihood (GTR + Felsenstein pruning) for MI455X / gfx1250.
//
// Roofline: tips = 205 MB (read once) -> ~9us @ 23.3 TB/s. Compute <1 GFLOP.
// Design: ONE fused pass; each wave owns 16 sites and walks the whole 511-node
// tree depth-first; internal partials live in registers + a 7-level per-wave
// LDS stack. Zero intermediate HBM traffic.
//
// Matrix op: V_WMMA_F32_16X16X4_F32.  D(16x16) = A(16x4) x B(4x16):
//   A rows 0..3 = P_edge (4x4), rows 4..15 = 0 (pre-swizzled per-lane in ws)
//   B cols      = 16 sites, rows = 4 states (child partial, transposed)
//   D[i,s]      = message, out-state i in VGPR row i, site s in lane s
// D re-enters the next level's B operand via one convergent ds_swizzle
// (SWAPX16, offset 0x401F) + v_cndmask — no full transpose needed.
// ---------------------------------------------------------------------------

#define NUM_TIPS   256
#define NUM_SITES  50000
#define N_NODES    511
#define N_EDGES    510
#define NGROUPS    (NUM_SITES / 16)   // 3125 site-groups of 16 (divides evenly)
#define WAVES_PER_BLOCK 8             // 256 threads = 8 wave32

typedef __attribute__((ext_vector_type(2))) float v2f;
typedef __attribute__((ext_vector_type(8))) float v8f;

// Workspace layout (floats):
//   [0..3]                       pi (softmax of pi_logits)
//   [128 .. 128+510*64)          Pexp: per edge, 64 floats = per-lane (A.v0,A.v1)
//   [128+510*64 .. +NGROUPS)     per-group partial log-likelihood sums
// Total ~144 KB.

// ---------------------------------------------------------------------------
// Kernel 1: GTR rate matrix + per-edge expm(Q*t), written pre-swizzled into
// the WMMA A-operand per-lane layout (lane i<4: P[i][0],P[i][1];
// lane 16+i: P[i][2],P[i][3]; all other lanes 0).
// ---------------------------------------------------------------------------
__global__ void gtr_prep_kernel(const float* __restrict__ edge_len,
                                const float* __restrict__ rates,
                                const float* __restrict__ pi_logits,
                                float* __restrict__ ws) {
  const int e = threadIdx.x;

  // pi = softmax(pi_logits)  (redundantly per thread; trivial)
  float pl[4];
#pragma unroll
  for (int i = 0; i < 4; ++i) pl[i] = pi_logits[i];
  float mx = fmaxf(fmaxf(pl[0], pl[1]), fmaxf(pl[2], pl[3]));
  float ex[4], ssum = 0.f;
#pragma unroll
  for (int i = 0; i < 4; ++i) { ex[i] = expf(pl[i] - mx); ssum += ex[i]; }
  float pi[4];
#pragma unroll
  for (int i = 0; i < 4; ++i) pi[i] = ex[i] / ssum;

  // softplus(rates)
  float r[6];
#pragma unroll
  for (int k = 0; k < 6; ++k) {
    float x = rates[k];
    r[k] = (x > 20.f) ? x : log1pf(expf(x));
  }

  // S (symmetric, zero diag), Q = S*diag(pi), zero row sums, mean rate 1
  float Ssym[4][4] = {{0.f,  r[0], r[1], r[2]},
                      {r[0], 0.f,  r[3], r[4]},
                      {r[1], r[3], 0.f,  r[5]},
                      {r[2], r[4], r[5], 0.f}};
  float Q[4][4];
#pragma unroll
  for (int i = 0; i < 4; ++i)
#pragma unroll
    for (int j = 0; j < 4; ++j) Q[i][j] = Ssym[i][j] * pi[j];
#pragma unroll
  for (int i = 0; i < 4; ++i) {
    float rs = 0.f;
#pragma unroll
    for (int j = 0; j < 4; ++j) rs += Q[i][j];
    Q[i][i] -= rs;
  }
  float mu = 0.f;
#pragma unroll
  for (int i = 0; i < 4; ++i) mu -= pi[i] * Q[i][i];
  float inv_mu = 1.0f / mu;
#pragma unroll
  for (int i = 0; i < 4; ++i)
#pragma unroll
    for (int j = 0; j < 4; ++j) Q[i][j] *= inv_mu;

  if (e == 0) {
    ws[0] = pi[0]; ws[1] = pi[1]; ws[2] = pi[2]; ws[3] = pi[3];
  }
  if (e >= N_EDGES) return;

  // M = Q * t, scaling + squaring with degree-12 Taylor (Horner)
  float t = edge_len[e];
  float M[4][4];
#pragma unroll
  for (int i = 0; i < 4; ++i)
#pragma unroll
    for (int j = 0; j < 4; ++j) M[i][j] = Q[i][j] * t;

  float nrm = 0.f;
#pragma unroll
  for (int i = 0; i < 4; ++i) {
    float rs = 0.f;
#pragma unroll
    for (int j = 0; j < 4; ++j) rs += fabsf(M[i][j]);
    nrm = fmaxf(nrm, rs);
  }
  int sc = 0;
  while (nrm > 0.25f && sc < 30) { nrm *= 0.5f; ++sc; }
  float scale = 1.0f / (float)(1u << sc);
#pragma unroll
  for (int i = 0; i < 4; ++i)
#pragma unroll
    for (int j = 0; j < 4; ++j) M[i][j] *= scale;

  float F[4][4];
#pragma unroll
  for (int i = 0; i < 4; ++i)
#pragma unroll
    for (int j = 0; j < 4; ++j) F[i][j] = (i == j) ? 1.f : 0.f;

  for (int k = 12; k >= 1; --k) {
    float G[4][4];
#pragma unroll
    for (int i = 0; i < 4; ++i)
#pragma unroll
      for (int j = 0; j < 4; ++j) {
        float s = 0.f;
#pragma unroll
        for (int q = 0; q < 4; ++q) s += M[i][q] * F[q][j];
        G[i][j] = s;
      }
    float ik = 1.0f / (float)k;
#pragma unroll
    for (int i = 0; i < 4; ++i)
#pragma unroll
      for (int j = 0; j < 4; ++j) F[i][j] = ((i == j) ? 1.f : 0.f) + G[i][j] * ik;
  }
  for (int q = 0; q < sc; ++q) {
    float G[4][4];
#pragma unroll
    for (int i = 0; i < 4; ++i)
#pragma unroll
      for (int j = 0; j < 4; ++j) {
        float s = 0.f;
#pragma unroll
        for (int p = 0; p < 4; ++p) s += F[i][p] * F[p][j];
        G[i][j] = s;
      }
#pragma unroll
    for (int i = 0; i < 4; ++i)
#pragma unroll
      for (int j = 0; j < 4; ++j) F[i][j] = G[i][j];
  }

  // Pre-swizzled A-operand store: [edge][lane][2]
  float* Pe = ws + 128 + (size_t)e * 64;
#pragma unroll
  for (int l = 0; l < 32; ++l) {
    float v0 = 0.f, v1 = 0.f;
    if (l < 4)                    { v0 = F[l][0];      v1 = F[l][1]; }
    else if (l >= 16 && l < 20)   { v0 = F[l - 16][2]; v1 = F[l - 16][3]; }
    Pe[2 * l]     = v0;
    Pe[2 * l + 1] = v1;
  }
}

// ---------------------------------------------------------------------------
// Convergent lane-half swap: lane l gets value from lane l^16.
// ds_swizzle group-of-32 encoding: xor_mask=0x10, or=0, and=0x1F -> 0x401F.
// Executed by ALL lanes (no EXEC games), unlike a sunk __shfl_xor.
// ---------------------------------------------------------------------------
__device__ __forceinline__ float swap16(float x) {
  return __int_as_float(__builtin_amdgcn_ds_swizzle(__float_as_int(x), 0x401F));
}

// ---------------------------------------------------------------------------
// One WMMA edge-message: m[i,s] = sum_j P[i,j] * L[j,s]  (16 sites per call)
// ---------------------------------------------------------------------------
__device__ __forceinline__ void edge_msg(const float* __restrict__ Pexp, int edge,
                                         float b0, float b1, int lane, float m[4]) {
  v2f a = *reinterpret_cast<const v2f*>(Pexp + (size_t)edge * 64 + 2 * lane);
  v2f b; b[0] = b0; b[1] = b1;
  v8f c = {};
  v8f d = __builtin_amdgcn_wmma_f32_16x16x4_f32(false, a, false, b,
                                                (short)0, c, false, false);
  m[0] = d[0]; m[1] = d[1]; m[2] = d[2]; m[3] = d[3];
}

// ---------------------------------------------------------------------------
// Kernel 2: fused whole-tree pruning. One wave = 16 sites through all 510
// edges via depth-first traversal; left-sibling messages in per-wave LDS stack.
// All branch conditions are wave-uniform => EXEC stays all-ones for WMMA.
// ---------------------------------------------------------------------------
__global__ void __launch_bounds__(256)
felsenstein_kernel(const float* __restrict__ tips,
                   const float* __restrict__ ws_pi,
                   const float* __restrict__ Pexp,
                   float* __restrict__ gsum) {
  __shared__ float4 stk[WAVES_PER_BLOCK][7][32];   // 28 KB: levels 1..7
  const int lane  = threadIdx.x & 31;
  const int wave  = threadIdx.x >> 5;
  const int group = blockIdx.x * WAVES_PER_BLOCK + wave;
  if (group >= NGROUPS) return;                    // wave-uniform exit

  const int base_elem = group * 16 * 4;            // 16 sites * 4 states
  // B-operand load offset: lane s<16 -> elems 4s,4s+1 ; lane 16+s -> 4s+2,4s+3
  const int loff = (lane < 16) ? (4 * lane) : (4 * (lane - 16) + 2);
  const float* tbase = tips + base_elem + loff;    // + tip * NUM_SITES*4

  float cur[4];
  for (int pair = 0; pair < 128; ++pair) {
    const float* pL = tbase + (size_t)(2 * pair) * (NUM_SITES * 4);
    // Prefetch next pair's tip rows (global_prefetch_b8) to overlap the
    // upcoming DFS merge work with the HBM stream.
    if (pair < 127) {
      __builtin_prefetch(pL + 2 * (size_t)(NUM_SITES * 4), 0, 0);
      __builtin_prefetch(pL + 3 * (size_t)(NUM_SITES * 4), 0, 0);
    }

    // --- two sibling tips -> level-7 partial ---
    float mL[4], mR[4];
    float2 Lp = *reinterpret_cast<const float2*>(pL);      // global_load_b64
    edge_msg(Pexp, 254 + 2 * pair, Lp.x, Lp.y, lane, mL);  // edge of tip 2*pair
    const float* pR = pL + NUM_SITES * 4;
    float2 Rp = *reinterpret_cast<const float2*>(pR);
    edge_msg(Pexp, 255 + 2 * pair, Rp.x, Rp.y, lane, mR);  // edge of tip 2*pair+1
    cur[0] = mL[0] * mR[0]; cur[1] = mL[1] * mR[1];
    cur[2] = mL[2] * mR[2]; cur[3] = mL[3] * mR[3];

    // --- merge upward while we are a right child ---
    int lev = 7, pos = pair;
    for (;;) {
      // repack partial (D layout) into next B operand: convergent half-swap
      float s2 = swap16(cur[2]);
      float s3 = swap16(cur[3]);
      float b0 = (lane < 16) ? cur[0] : s2;
      float b1 = (lane < 16) ? cur[1] : s3;
      int node = (1 << lev) - 1 + pos;
      float m[4];
      edge_msg(Pexp, node - 1, b0, b1, lane, m);
      if ((pos & 1) == 0) {                        // left child: park message
        stk[wave][lev - 1][lane] = make_float4(m[0], m[1], m[2], m[3]);
        break;
      }
      float4 sib = stk[wave][lev - 1][lane];       // right child: combine
      cur[0] = sib.x * m[0]; cur[1] = sib.y * m[1];
      cur[2] = sib.z * m[2]; cur[3] = sib.w * m[3];
      --lev; pos >>= 1;
      if (lev == 0) break;                         // reached root (pair==127)
    }
  }

  // root partial in cur (lanes 0..15 = 16 sites)
  float lik = ws_pi[0] * cur[0] + ws_pi[1] * cur[1] +
              ws_pi[2] * cur[2] + ws_pi[3] * cur[3];
  float l = (lane < 16) ? logf(lik) : 0.0f;
  // wave sum: all lanes participate (lanes >=16 contribute 0)
#pragma unroll
  for (int off = 16; off >= 1; off >>= 1) l += __shfl_xor(l, off, 32);
  if (lane == 0) gsum[group] = l;
}

// ---------------------------------------------------------------------------
// Kernel 3: deterministic fixed-order reduction of the 3125 group sums.
// ---------------------------------------------------------------------------
__global__ void final_reduce_kernel(const float* __restrict__ gsum,
                                    float* __restrict__ out) {
  __shared__ float sh[256];
  float acc = 0.f;
  for (int i = threadIdx.x; i < NGROUPS; i += 256) acc += gsum[i];
  sh[threadIdx.x] = acc;
  __syncthreads();
  for (int s = 128; s > 0; s >>= 1) {
    if ((int)threadIdx.x < s) sh[threadIdx.x] += sh[threadIdx.x + s];
    __syncthreads();
  }
  if (threadIdx.x == 0) out[0] = sh[0];
}

// ---------------------------------------------------------------------------
extern "C" void kernel_launch(void* const* d_in, const int* in_sizes, int n_in,
                              void* d_out, int out_size, void* d_ws, size_t ws_size,
                              hipStream_t stream) {
  (void)in_sizes; (void)n_in; (void)out_size; (void)ws_size;
  const float* edge_len = (const float*)d_in[0];   // (510,)
  const float* tips     = (const float*)d_in[1];   // (256, 50000, 4)
  const float* rates    = (const float*)d_in[2];   // (6,)
  const float* pil      = (const float*)d_in[3];   // (4,)

  float* ws   = (float*)d_ws;
  float* pi   = ws;                                // 4 floats
  float* Pexp = ws + 128;                          // 510*64 floats (swizzled P)
  float* gsum = ws + 128 + N_EDGES * 64;           // NGROUPS floats

  gtr_prep_kernel<<<1, 512, 0, stream>>>(edge_len, rates, pil, ws);

  int blocks = (NGROUPS + WAVES_PER_BLOCK - 1) / WAVES_PER_BLOCK;
  felsenstein_kernel<<<blocks, 32 * WAVES_PER_BLOCK, 0, stream>>>(tips, pi, Pexp, gsum);

  final_reduce_kernel<<<1, 256, 0, stream>>>(gsum, (float*)d_out);
}